// DilatedAttention_15625091023338
// MI455X (gfx1250) — compile-verified
//
#include <hip/hip_runtime.h>
#include <math.h>

// Problem constants (match reference)
#define NB   8
#define CC   64
#define HH   256
#define WW   256
#define GG   8
#define K2   9
#define ROWS 8          // output rows per block in stencil kernel
#define BN_EPS 1e-5f

typedef __attribute__((ext_vector_type(2))) float v2f;
typedef __attribute__((ext_vector_type(8))) float v8f;

// ---------------------------------------------------------------------------
// Kernel 1: pooled[n*C+c] = mean over H*W of x[n,c,:,:]
// 512 blocks x 256 threads; float4 (b128) coalesced loads; LDS tree reduce.
// ---------------------------------------------------------------------------
__global__ void pool_kernel(const float* __restrict__ x, float* __restrict__ pooled) {
    const int nc = blockIdx.x;                       // 0..511
    const float4* __restrict__ x4 =
        (const float4*)(x + (size_t)nc * (HH * WW)); // 16384 float4 per channel
    float s = 0.f;
    for (int i = threadIdx.x; i < (HH * WW) / 4; i += 256) {
        float4 v = x4[i];
        s += v.x + v.y + v.z + v.w;
    }
    __shared__ float red[256];
    red[threadIdx.x] = s;
    __syncthreads();
    for (int off = 128; off > 0; off >>= 1) {
        if ((int)threadIdx.x < off) red[threadIdx.x] += red[threadIdx.x + off];
        __syncthreads();
    }
    if (threadIdx.x == 0) pooled[nc] = red[0] * (1.f / (float)(HH * WW));
}

// ---------------------------------------------------------------------------
// Kernel 2: lf = tanh(BN(pooled @ conv_w^T)) via native fp32 WMMA.
// One wave (32 threads). D = A(16x4) x B(4x16) + C(16x16), fp32 throughout.
//   M = batch (8 used of 16), N = output channel (72 used of 80), K = C = 64.
// A layout (32-bit A 16x4): lanes 0-15 -> M=lane, VGPR{0,1} = K={k0,k0+1};
//                           lanes 16-31 -> M=lane-16, VGPR{0,1} = K={k0+2,k0+3}.
// B layout mirrors A with N in place of M (row of B striped across lanes).
// C/D layout: VGPR r holds row r (lanes 0-15) / row r+8 (lanes 16-31), col=lane&15.
// All masking done with cndmask-style selects (no EXEC divergence before WMMA).
// ---------------------------------------------------------------------------
__global__ void lf_kernel(const float* __restrict__ pooled,   // (8,64)
                          const float* __restrict__ conv_w,   // (72,64)
                          const float* __restrict__ gamma,
                          const float* __restrict__ beta,
                          const float* __restrict__ mean,
                          const float* __restrict__ var,
                          float* __restrict__ lf) {            // (8,72)
    const int lane = threadIdx.x;        // 0..31
    const int half = lane >> 4;          // K sub-pair select
    const int l15  = lane & 15;

    const int   arow  = l15;                       // M index 0..15
    const float amask = (arow < NB) ? 1.f : 0.f;   // rows 8..15 of A are zero pad
    const int   ar    = arow & (NB - 1);           // clamped row for safe load

    for (int nt = 0; nt < 5; ++nt) {
        const int   col   = nt * 16 + l15;                 // output channel 0..79
        const float bmask = (col < GG * K2) ? 1.f : 0.f;   // cols 72..79 zero pad
        const int   bc    = (col < GG * K2) ? col : 0;     // clamped for safe load

        v8f acc = {};
        for (int k0 = 0; k0 < CC; k0 += 4) {
            const int ka = k0 + half * 2;
            v2f a, b;
            a.x = pooled[ar * CC + ka]     * amask;
            a.y = pooled[ar * CC + ka + 1] * amask;
            b.x = conv_w[bc * CC + ka]     * bmask;       // B[k][n] = conv_w[n][k]
            b.y = conv_w[bc * CC + ka + 1] * bmask;
            acc = __builtin_amdgcn_wmma_f32_16x16x4_f32(
                /*neg_a=*/false, a, /*neg_b=*/false, b,
                /*c_mod=*/(short)0, acc, /*reuse_a=*/false, /*reuse_b=*/false);
        }

        // Epilogue: eval-mode BN + tanh, store valid (M<8, col<72) elements.
        if (col < GG * K2) {
            const float sc = gamma[col] * rsqrtf(var[col] + BN_EPS);
            const float mu = mean[col];
            const float bt = beta[col];
#pragma unroll
            for (int r = 0; r < 8; ++r) {
                const int M = r + half * 8;           // row held by this vgpr/lane
                if (M < NB) lf[M * (GG * K2) + col] = tanhf((acc[r] - mu) * sc + bt);
            }
        }
    }
}

// ---------------------------------------------------------------------------
// Kernel 3: per-pixel dynamic 3x3 stencil + DSA epilogue. HBM-bandwidth bound.
// Block = 256 threads handles an 8-row strip of one (n,c) plane.
// 10 rows (8 + reflect halo) staged in LDS (10 KB) -> 1.25x read amplification,
// fully coalesced loads/stores. Reflection (numpy 'reflect', no edge dup):
// idx -1 -> 1, idx H -> H-2.
// ---------------------------------------------------------------------------
__global__ void stencil_kernel(const float* __restrict__ x,
                               const float* __restrict__ pooled,     // (8,64) means
                               const float* __restrict__ lf,         // (8,72)
                               const float* __restrict__ lamb_l,     // (64)
                               const float* __restrict__ lamb_h,     // (64)
                               const float* __restrict__ inside_all, // (64)
                               float* __restrict__ out) {
    const int w  = threadIdx.x;           // 0..255 == full row
    const int h0 = blockIdx.x * ROWS;     // strip start row
    const int c  = blockIdx.y;
    const int n  = blockIdx.z;
    const int g  = c >> 3;                // group = c / (C/G)

    const size_t plane = (size_t)(n * CC + c) * (HH * WW);
    const float* __restrict__ xc = x + plane;
    float* __restrict__       oc = out + plane;

    __shared__ float tile[ROWS + 2][WW];

    // Stage 10 rows with H-reflection at tile edges.
#pragma unroll
    for (int r = 0; r < ROWS + 2; ++r) {
        int hg = h0 - 1 + r;
        hg = (hg < 0) ? -hg : ((hg >= HH) ? (2 * HH - 2 - hg) : hg);
        tile[r][w] = xc[hg * WW + w];
    }
    __syncthreads();

    // Per-block uniform parameters.
    float f[K2];
#pragma unroll
    for (int t = 0; t < K2; ++t) f[t] = lf[n * (GG * K2) + g * K2 + t];
    const float gap = pooled[n * CC + c];
    const float ia  = inside_all[c];
    const float ll  = lamb_l[c];
    const float lh1 = lamb_h[c] + 1.f;
    const float ia1 = ia + 1.f;

    // W-reflection indices for this lane.
    const int wm = (w == 0) ? 1 : (w - 1);
    const int wp = (w == WW - 1) ? (WW - 2) : (w + 1);

#pragma unroll
    for (int r = 0; r < ROWS; ++r) {
        // tap t = i*3+j corresponds to x[h+i-1, w+j-1]; tile row r is h0+r-1.
        float acc = tile[r    ][wm] * f[0] + tile[r    ][w] * f[1] + tile[r    ][wp] * f[2]
                  + tile[r + 1][wm] * f[3] + tile[r + 1][w] * f[4] + tile[r + 1][wp] * f[5]
                  + tile[r + 2][wm] * f[6] + tile[r + 2][w] * f[7] + tile[r + 2][wp] * f[8];
        const float xv  = tile[r + 1][w];
        const float res = (acc * ia1 - ia * gap) * ll + xv * lh1;
        oc[(h0 + r) * WW + w] = res;
    }
}

// ---------------------------------------------------------------------------
// Launch: pool -> lf (WMMA) -> stencil, all on `stream`. Workspace layout:
//   ws[0..511]    pooled (8x64 means)
//   ws[512..1087] lf (8x72)
// ---------------------------------------------------------------------------
extern "C" void kernel_launch(void* const* d_in, const int* in_sizes, int n_in,
                              void* d_out, int out_size, void* d_ws, size_t ws_size,
                              hipStream_t stream) {
    (void)in_sizes; (void)n_in; (void)out_size; (void)ws_size;

    const float* x          = (const float*)d_in[0];
    const float* conv_w     = (const float*)d_in[1];
    const float* bn_gamma   = (const float*)d_in[2];
    const float* bn_beta    = (const float*)d_in[3];
    const float* bn_mean    = (const float*)d_in[4];
    const float* bn_var     = (const float*)d_in[5];
    const float* lamb_l     = (const float*)d_in[6];
    const float* lamb_h     = (const float*)d_in[7];
    const float* inside_all = (const float*)d_in[8];
    float*       out        = (float*)d_out;

    float* pooled = (float*)d_ws;          // 512 floats
    float* lf     = pooled + NB * CC;      // 576 floats

    pool_kernel<<<NB * CC, 256, 0, stream>>>(x, pooled);
    lf_kernel<<<1, 32, 0, stream>>>(pooled, conv_w, bn_gamma, bn_beta,
                                    bn_mean, bn_var, lf);
    stencil_kernel<<<dim3(HH / ROWS, CC, NB), WW, 0, stream>>>(
        x, pooled, lf, lamb_l, lamb_h, inside_all, out);
}